// CapsuleLayer_58600533786884
// MI455X (gfx1250) — compile-verified
//
#include <hip/hip_runtime.h>
#include <math.h>

typedef float v2f __attribute__((ext_vector_type(2)));
typedef float v8f __attribute__((ext_vector_type(8)));

#define KK      3
#define CI      32
#define HH      14
#define CIN     8
#define NN      288      // CI*KK*KK
#define DD      16
#define WOUT    12
#define BATCH   4
#define SITES_PER_WG 16
#define NTILES  36       // (BATCH*WOUT*WOUT)/16
#define AA      32
#define RITERS  3

// padded n-stride: 260 mod 64 = 4 -> rotates LDS bank groups per n, killing the
// 16-way conflict on per-thread float4 d-row reads in the logits update, while
// the softmax-loop reads (32 consecutive words/wave) stay conflict-free.
#define NSTRIDE 260

// LDS layout (floats):
//   priors: [NN][NSTRIDE]  (within n: site*16 + d)  74880 floats
//   logits: [16][NN]                                 4608 floats
//   outv  : [16][16]                                  256 floats
#define PRIORS_OFF 0
#define LOGITS_OFF (NN * NSTRIDE)
#define OUTV_OFF   (LOGITS_OFF + 16 * NN)
#define SMEM_FLOATS (OUTV_OFF + 256)   // 79744 floats = 318976 B < 320KB

// one-instruction XOR butterfly within 16-lane groups via DS_SWIZZLE_B32
// group-of-32 mode: lane' = ((lane & and) | or) ^ xor ; offset = xor<<10|or<<5|and
template <int XMASK>
__device__ __forceinline__ float swz_xor(float v) {
    return __int_as_float(
        __builtin_amdgcn_ds_swizzle(__float_as_int(v), (XMASK << 10) | 0x1f));
}
__device__ __forceinline__ float red16_add(float v) {
    v += swz_xor<1>(v); v += swz_xor<2>(v); v += swz_xor<4>(v); v += swz_xor<8>(v);
    return v;
}
__device__ __forceinline__ float red16_max(float v) {
    v = fmaxf(v, swz_xor<1>(v)); v = fmaxf(v, swz_xor<2>(v));
    v = fmaxf(v, swz_xor<4>(v)); v = fmaxf(v, swz_xor<8>(v));
    return v;
}

__global__ __launch_bounds__(256, 1)
void capsule_route_kernel(const float* __restrict__ x,
                          const float* __restrict__ rw,
                          const float* __restrict__ bias,
                          float* __restrict__ out)
{
    extern __shared__ float smem[];
    float* sh_pri = smem + PRIORS_OFF;
    float* sh_log = smem + LOGITS_OFF;
    float* sh_out = smem + OUTV_OFF;

    const int tid  = threadIdx.x;
    const int a    = blockIdx.y;   // output-capsule group 0..31
    const int tile = blockIdx.x;   // 16-site tile 0..35

    // zero logits (read by iteration-1 softmax; accumulated by iter-0 update)
    for (int idx = tid; idx < 16 * NN; idx += 256) sh_log[idx] = 0.0f;

    // ---------------- Phase 1: priors via V_WMMA_F32_16X16X4_F32 ----------------
    {
        const int lane = tid & 31;
        const int wave = tid >> 5;     // 0..7
        const int m    = lane & 15;    // A-matrix row (site) / C-matrix column (d)
        const int l16  = lane >> 4;    // 0/1 -> K half
        const int c0   = 2 * l16;      // first K index in this lane's VGPR0

        // site for this lane's A-matrix row
        const int sgA = tile * SITES_PER_WG + m;
        const int bA  = sgA / 144;
        const int rA  = sgA % 144;
        const int iA  = rA / WOUT;
        const int jA  = rA % WOUT;

        // x[b][ci][iA+ki][jA+kj][c]: ci stride = HH*HH*CIN = 1568
        const float* xsite = x + ((size_t)bA * CI * HH * HH + (size_t)iA * HH + jA) * CIN;
        const float* wa    = rw + (size_t)a * NN * 128 + m;    // rw[a][n][c][d=m]

        // each wave owns ci = wave, wave+8, ... (4 values -> 36 n each)
        for (int ci = wave; ci < CI; ci += 8) {
            const float* xc   = xsite + (size_t)ci * (HH * HH * CIN);
            const float* wc   = wa    + (size_t)ci * 9 * 128;
            float*       dstc = sh_pri + (ci * 9) * NSTRIDE + (l16 * 8) * 16 + m;

            if (ci + 8 < CI) {  // wave-uniform guard; speculative prefetch of next weights
                __builtin_prefetch(wc + 8 * 9 * 128, 0, 3);
            }

            #pragma unroll
            for (int ki = 0; ki < KK; ++ki) {
                #pragma unroll
                for (int kj = 0; kj < KK; ++kj) {
                    const float* xp = xc + (ki * HH + kj) * CIN;
                    const float* wp = wc + (ki * KK + kj) * 128;

                    v2f A0, A1;                       // 16x8 A -> two 16x4 blocks
                    A0.x = xp[c0];     A0.y = xp[c0 + 1];
                    A1.x = xp[c0 + 4]; A1.y = xp[c0 + 5];
                    v2f B0, B1;                       // 8x16 B -> two 4x16 blocks
                    B0.x = wp[(c0    ) * 16]; B0.y = wp[(c0 + 1) * 16];
                    B1.x = wp[(c0 + 4) * 16]; B1.y = wp[(c0 + 5) * 16];

                    v8f acc = {0.f, 0.f, 0.f, 0.f, 0.f, 0.f, 0.f, 0.f};
                    acc = __builtin_amdgcn_wmma_f32_16x16x4_f32(false, A0, false, B0,
                                                                (short)0, acc, false, false);
                    acc = __builtin_amdgcn_wmma_f32_16x16x4_f32(false, A1, false, B1,
                                                                (short)0, acc, false, false);

                    float* dst = dstc + (ki * KK + kj) * NSTRIDE;
                    #pragma unroll
                    for (int v = 0; v < 8; ++v) dst[v * 16] = acc[v];
                }
            }
        }
    }
    __syncthreads();

    // ---------------- Phase 2: dynamic routing entirely in LDS ----------------
    const int sl = tid >> 4;   // site-local 0..15
    const int d  = tid & 15;   // capsule output dim 0..15

    const int sg = tile * SITES_PER_WG + sl;
    const int b  = sg / 144;
    const int r  = sg % 144;
    const int i0 = r / WOUT;
    const int j0 = r % WOUT;

    const float bv = bias[a * DD + d];
    const float* pri_sd = sh_pri + sl * 16 + d;  // stride NSTRIDE over n
    float out_reg = 0.f;

    for (int it = 0; it < RITERS; ++it) {
        float sv;
        if (it == 0) {
            // softmax of all-zero logits == uniform 1/NN
            float acc = 0.f;
            for (int n = 0; n < NN; ++n) acc += pri_sd[n * NSTRIDE];
            sv = acc * (1.0f / NN) + bv;
        } else {
            // softmax over n of sh_log[sl][*]: 16 threads per site cooperate
            const float* lg = sh_log + sl * NN;
            float mx = -3.0e38f;
            for (int k = 0; k < NN / 16; ++k) mx = fmaxf(mx, lg[d + 16 * k]);
            mx = red16_max(mx);
            float se = 0.f;
            for (int k = 0; k < NN / 16; ++k) se += __expf(lg[d + 16 * k] - mx);
            se = red16_add(se);
            const float inv = 1.0f / se;

            float acc = 0.f;
            for (int n = 0; n < NN; ++n) {
                const float p = __expf(lg[n] - mx) * inv;   // LDS broadcast read
                acc += p * pri_sd[n * NSTRIDE];
            }
            sv = acc + bv;
        }

        // squash: sn = sum_d s^2 ; out = s * sqrt(sn)/(1+sn)
        const float sq = red16_add(sv * sv);
        const float scale = sqrtf(sq) / (1.0f + sq);
        out_reg = sv * scale;
        sh_out[sl * 16 + d] = out_reg;
        __syncthreads();

        if (it != RITERS - 1) {
            // logits[sl][n] += dot_d(priors[n][sl][:], out[sl][:])
            const float4* ov = (const float4*)(sh_out + sl * 16);
            const float4 o0 = ov[0], o1 = ov[1], o2 = ov[2], o3 = ov[3];
            for (int k = 0; k < NN / 16; ++k) {
                const int n = d + 16 * k;
                const float4* pv = (const float4*)(sh_pri + n * NSTRIDE + sl * 16);
                const float4 p0 = pv[0], p1 = pv[1], p2 = pv[2], p3 = pv[3];
                const float dot =
                    p0.x * o0.x + p0.y * o0.y + p0.z * o0.z + p0.w * o0.w +
                    p1.x * o1.x + p1.y * o1.y + p1.z * o1.z + p1.w * o1.w +
                    p2.x * o2.x + p2.y * o2.y + p2.z * o2.z + p2.w * o2.w +
                    p3.x * o3.x + p3.y * o3.y + p3.z * o3.z + p3.w * o3.w;
                sh_log[sl * NN + n] += dot;
            }
            __syncthreads();
        }
    }

    // output layout: out[b][a][i][j][d]  (reference transposes axes 0/1)
    out[((((size_t)b * AA + a) * WOUT + i0) * WOUT + j0) * DD + d] = out_reg;
}

extern "C" void kernel_launch(void* const* d_in, const int* in_sizes, int n_in,
                              void* d_out, int out_size, void* d_ws, size_t ws_size,
                              hipStream_t stream) {
    (void)in_sizes; (void)n_in; (void)out_size; (void)d_ws; (void)ws_size;
    const float* x    = (const float*)d_in[0];
    const float* rw   = (const float*)d_in[1];
    const float* bias = (const float*)d_in[2];
    float* out = (float*)d_out;

    dim3 grid(NTILES, AA);   // 36 x 32 = 1152 workgroups, one per (a, site-tile)
    dim3 block(256);         // 8 waves (wave32)
    size_t shmem = SMEM_FLOATS * sizeof(float);  // 318976 bytes
    capsule_route_kernel<<<grid, block, shmem, stream>>>(x, rw, bias, out);
}